// QuantumEncodingLayer_62405874811468
// MI455X (gfx1250) — compile-verified
//
#include <hip/hip_runtime.h>
#include <hip/hip_bf16.h>

typedef __attribute__((ext_vector_type(2))) float v2f;
typedef __attribute__((ext_vector_type(8))) float v8f;

#define DIM        64
#define KCHEB      160      // Chebyshev terms (multiple of 4 for K-steps)
#define MSTART     208      // Miller recurrence start index (> KCHEB + margin)
#define RHO        16.25f   // spectral radius bound of -i(adag - a), dim 64
#define BATCH_TILE 64
#define NBATCH     16384

// ---------------------------------------------------------------------------
// Kernel A: batch-independent Chebyshev Krylov vectors
//   phi_0 = psi0; phi_1 = -(1/rho) G phi_0; phi_{k+1} = -(2/rho) G phi_k + phi_{k-1}
//   (G phi)[i] = sqrt(i) phi[i-1] - sqrt(i+1) phi[i+1]   (G = adag - a, tridiagonal)
// phi stored row-major [DIM][KCHEB] so WMMA A-fragments load (k,k+1) pairs as b64.
// ---------------------------------------------------------------------------
__global__ __launch_bounds__(DIM) void cheb_phi_kernel(
        const float* __restrict__ psi0, float* __restrict__ phi) {
    __shared__ float cur[DIM];
    const int i  = threadIdx.x;
    const float si  = sqrtf((float)i);
    const float si1 = sqrtf((float)(i + 1));

    float prev = psi0[i];                    // phi_0
    phi[i * KCHEB + 0] = prev;
    cur[i] = prev;
    __syncthreads();
    float lo = (i > 0)       ? cur[i - 1] : 0.f;
    float hi = (i < DIM - 1) ? cur[i + 1] : 0.f;
    float c  = -(1.f / RHO) * (si * lo - si1 * hi);   // phi_1
    phi[i * KCHEB + 1] = c;

    for (int k = 2; k < KCHEB; ++k) {
        __syncthreads();
        cur[i] = c;
        __syncthreads();
        lo = (i > 0)       ? cur[i - 1] : 0.f;
        hi = (i < DIM - 1) ? cur[i + 1] : 0.f;
        float nx = -(2.f / RHO) * (si * lo - si1 * hi) + prev;
        prev = c;
        c = nx;
        phi[i * KCHEB + k] = c;
    }
}

// ---------------------------------------------------------------------------
// Kernel B: per-batch Bessel coefficients (Miller downward recurrence) into
// LDS, then Y[64 x 64cols] = Phi[64 x K] @ W[K x 64cols] via WMMA f32 16x16x4.
// Output written as complex64 (imag = 0).
// ---------------------------------------------------------------------------
__global__ __launch_bounds__(BATCH_TILE) void displace_kernel(
        const float* __restrict__ alpha,
        const float* __restrict__ phi,
        float2* __restrict__ out) {
    __shared__ float W[KCHEB * BATCH_TILE];   // [k][col], 40 KB
    const int tid   = threadIdx.x;
    const int bBase = blockIdx.x * BATCH_TILE;

    // ---- Phase 1: coefficients w_k = (2 - d_k0) * J_k(|alpha|*rho) * s^k ----
    {
        const float a  = alpha[bBase + tid];
        float th = fmaxf(fabsf(a) * RHO, 1e-4f);
        const float s = (a >= 0.f) ? -1.f : 1.f;   // sign factor: (-1)^k folded with J_k parity
        float jp = 0.f, jc = 1e-30f, sum = 0.f;
        for (int k = MSTART; k >= 1; --k) {
            if (k < KCHEB) W[k * BATCH_TILE + tid] = jc;
            if ((k & 1) == 0) sum += 2.f * jc;                 // normalization terms
            float jm = (2.f * (float)k / th) * jc - jp;        // J_{k-1}
            jp = jc; jc = jm;
            if (fabsf(jc) > 1e18f) {                           // overflow rescale
                jc *= 1e-18f; jp *= 1e-18f; sum *= 1e-18f;
                const int j0 = (k < KCHEB) ? k : KCHEB;
                for (int j = j0; j < KCHEB; ++j) W[j * BATCH_TILE + tid] *= 1e-18f;
            }
        }
        W[0 * BATCH_TILE + tid] = jc;                          // J_0 (unnormalized)
        sum += jc;                                             // norm = J0 + 2*sum_even
        const float inv = 1.f / sum;
        float sk = 1.f, two = 1.f;
        for (int k = 0; k < KCHEB; ++k) {
            W[k * BATCH_TILE + tid] *= inv * two * sk;
            sk *= s; two = 2.f;
        }
    }
    __syncthreads();

    // ---- Phase 2: WMMA f32 16x16x4 GEMM ----
    const int lane = tid & 31;
    const int wave = tid >> 5;         // 2 waves: cols [0,32) and [32,64)
    const int hi   = lane >> 4;        // K-half select per ISA A/B layout
    const int lrow = lane & 15;
    const int colW = wave * 32;

    v8f acc[2][4];
    #pragma unroll
    for (int n = 0; n < 2; ++n)
        #pragma unroll
        for (int m = 0; m < 4; ++m) acc[n][m] = {};

    for (int k0 = 0; k0 < KCHEB; k0 += 4) {
        // A fragments: 16x4 f32. VGPR0 = K {k0 | k0+2}, VGPR1 = K {k0+1 | k0+3}
        v2f afr[4];
        #pragma unroll
        for (int m = 0; m < 4; ++m) {
            const int row = m * 16 + lrow;
            const float2 v = *reinterpret_cast<const float2*>(
                phi + row * KCHEB + k0 + 2 * hi);
            afr[m].x = v.x; afr[m].y = v.y;
        }
        // B fragments: 4x16 f32, row (K) striped across VGPRs, cols across lanes
        v2f bfr[2];
        #pragma unroll
        for (int n = 0; n < 2; ++n) {
            const int col = colW + n * 16 + lrow;
            bfr[n].x = W[(k0 + 2 * hi)     * BATCH_TILE + col];
            bfr[n].y = W[(k0 + 1 + 2 * hi) * BATCH_TILE + col];
        }
        #pragma unroll
        for (int n = 0; n < 2; ++n)
            #pragma unroll
            for (int m = 0; m < 4; ++m)
                acc[n][m] = __builtin_amdgcn_wmma_f32_16x16x4_f32(
                    false, afr[m], false, bfr[n],
                    (short)0, acc[n][m], false, false);
    }

    // ---- Store complex64: C/D layout -> row = m*16 + r + 8*hi, col = lane%16
    #pragma unroll
    for (int n = 0; n < 2; ++n) {
        const int b = bBase + colW + n * 16 + lrow;
        #pragma unroll
        for (int m = 0; m < 4; ++m) {
            #pragma unroll
            for (int r = 0; r < 8; ++r) {
                const int row = m * 16 + r + 8 * hi;
                out[(size_t)b * DIM + row] = make_float2(acc[n][m][r], 0.f);
            }
        }
    }
}

extern "C" void kernel_launch(void* const* d_in, const int* in_sizes, int n_in,
                              void* d_out, int out_size, void* d_ws, size_t ws_size,
                              hipStream_t stream) {
    const float* alpha = (const float*)d_in[0];   // [16384] f32
    const float* psi0  = (const float*)d_in[1];   // [64]    f32
    float* phi = (float*)d_ws;                    // 64*160*4 = 40 KB scratch

    cheb_phi_kernel<<<1, DIM, 0, stream>>>(psi0, phi);
    displace_kernel<<<NBATCH / BATCH_TILE, BATCH_TILE, 0, stream>>>(
        alpha, phi, (float2*)d_out);
}